// QuanvolutionGen256_65481071410599
// MI455X (gfx1250) — compile-verified
//
#include <hip/hip_runtime.h>
#include <math.h>

// CDNA5 fp32 WMMA operand types
typedef __attribute__((ext_vector_type(2))) float v2f;
typedef __attribute__((ext_vector_type(8))) float v8f;

struct Masks16 { unsigned int m[16]; };

// ---------------------------------------------------------------------------
// Kernel 1: quantum feature map, reduced analytically:
//   qfeat[m,i] = prod_{j : A[i,j]==1} cos(pix[m,j]/255*pi + theta[j])
// One block = 16 rows x 16 features. Cosines staged in LDS.
// ---------------------------------------------------------------------------
__global__ void qfeat_kernel(const float* __restrict__ x,
                             const float* __restrict__ theta,
                             float* __restrict__ qfeat,
                             Masks16 masks)
{
    __shared__ float cosv[16][17];           // +1 pad: no bank conflicts
    const int tid = threadIdx.x;
    const int rl  = tid >> 4;                // local row 0..15
    const int j   = tid & 15;                // pixel / feature index
    const int m   = blockIdx.x * 16 + rl;    // global row 0..1023 (= b*4 + p)
    const int b   = m >> 2;
    const int p   = m & 3;
    const int row = (p >> 1) * 16;           // p1*16 (always < 28)
    const int col = (p & 1) * 16 + j;        // p2*16 + j (zero-padded >= 28)
    const float val = (col < 28) ? x[b * 784 + row * 28 + col] : 0.0f;
    const float ang = val * (float)(3.14159265358979323846 / 255.0) + theta[j];
    cosv[rl][j] = cosf(ang);
    __syncthreads();

    const unsigned int mask = masks.m[j];    // GF(2) row for output feature j
    float prod = 1.0f;
    #pragma unroll
    for (int jj = 0; jj < 16; ++jj) {
        const float f = cosv[rl][jj];
        prod *= (((mask >> jj) & 1u) ? f : 1.0f);   // branchless select
    }
    qfeat[m * 16 + j] = prod;
}

// ---------------------------------------------------------------------------
// Generic fp32 WMMA GEMM:  D[M,N] = alpha * (A[M,K] @ W[N,K]^T + bias[N])
// One wave32 per 16x16 output tile, K stepped by 4 with V_WMMA_F32_16X16X4_F32.
// A-operand layout (ISA 7.12.2): lane&15 = row M, K pairs split across lane
// halves (lanes 0-15: K+0/K+1, lanes 16-31: K+2/K+3). B symmetric (lane = N).
// C/D: VGPR r holds rows r (lanes 0-15) and r+8 (lanes 16-31), N = lane&15.
// ---------------------------------------------------------------------------
__global__ void gemm_wmma(const float* __restrict__ A,
                          const float* __restrict__ W,
                          const float* __restrict__ bias,
                          float* __restrict__ D,
                          int N, int K, float alpha)
{
    const int tilesN = N >> 4;
    const int tm   = blockIdx.x / tilesN;
    const int tn   = blockIdx.x % tilesN;
    const int lane = threadIdx.x & 31;
    const int half = lane >> 4;              // which K-pair this lane holds
    const int l16  = lane & 15;

    const float* Arow = A + (size_t)(tm * 16 + l16) * K + half * 2;
    const float* Wrow = W + (size_t)(tn * 16 + l16) * K + half * 2;

    v8f acc = {0.f, 0.f, 0.f, 0.f, 0.f, 0.f, 0.f, 0.f};
    for (int k = 0; k < K; k += 4) {
        const v2f a = *(const v2f*)(Arow + k);   // A[row][k+half*2 .. +1]
        const v2f w = *(const v2f*)(Wrow + k);   // B[k+half*2 .. +1][col]
        acc = __builtin_amdgcn_wmma_f32_16x16x4_f32(
            /*neg_a=*/false, a, /*neg_b=*/false, w,
            /*c_mod=*/(short)0, acc, /*reuse_a=*/false, /*reuse_b=*/false);
    }

    const int colg = tn * 16 + l16;
    const float bv = bias[colg];
    #pragma unroll
    for (int r = 0; r < 8; ++r) {
        const int rowg = tm * 16 + r + half * 8;
        D[(size_t)rowg * N + colg] = alpha * (acc[r] + bv);
    }
}

// ---------------------------------------------------------------------------
// Kernel 4: attention over 4 tokens, 8 heads, D=32. One wave32 per (b,h),
// lane = head dim. Scores via xor-butterfly wave reduction, exact softmax.
// qkv row layout: (b*4+t)*768 ; q at +0, k at +256, v at +512, col = h*32+d.
// ---------------------------------------------------------------------------
__global__ void attn_kernel(const float* __restrict__ qkv,
                            float* __restrict__ o)
{
    const int lane = threadIdx.x & 31;
    const int wave = threadIdx.x >> 5;
    const int hh   = blockIdx.x * 8 + wave;      // 0..2047
    const int b    = hh >> 3;
    const int h    = hh & 7;

    const float* base = qkv + (size_t)b * 4 * 768 + h * 32 + lane;
    float q[4], k[4], v[4];
    #pragma unroll
    for (int t = 0; t < 4; ++t) {
        q[t] = base[t * 768 + 0];
        k[t] = base[t * 768 + 256];
        v[t] = base[t * 768 + 512];
    }

    const float scale = 0.17677669529663689f;    // 1/sqrt(32)
    float att[4][4];
    #pragma unroll
    for (int i = 0; i < 4; ++i) {
        #pragma unroll
        for (int jj = 0; jj < 4; ++jj) {
            float s = q[i] * k[jj];
            #pragma unroll
            for (int off = 16; off > 0; off >>= 1)
                s += __shfl_xor(s, off, 32);     // wave32 butterfly
            att[i][jj] = s * scale;
        }
    }

    float* orow = o + (size_t)b * 4 * 256 + h * 32 + lane;
    #pragma unroll
    for (int i = 0; i < 4; ++i) {
        const float mx = fmaxf(fmaxf(att[i][0], att[i][1]),
                               fmaxf(att[i][2], att[i][3]));
        const float e0 = expf(att[i][0] - mx);
        const float e1 = expf(att[i][1] - mx);
        const float e2 = expf(att[i][2] - mx);
        const float e3 = expf(att[i][3] - mx);
        const float inv = 1.0f / (e0 + e1 + e2 + e3);
        orow[i * 256] = (e0 * v[0] + e1 * v[1] + e2 * v[2] + e3 * v[3]) * inv;
    }
}

// ---------------------------------------------------------------------------
// Kernel 6: classifier + log_softmax. One wave32 per sample; lane-parallel K,
// xor-butterfly reduction of the 10 logits, exact logsumexp.
// flat[b] is simply oproj + b*1024 (contiguous (B,4,E) view).
// ---------------------------------------------------------------------------
__global__ void cls_kernel(const float* __restrict__ flat,   // (256,1024)
                           const float* __restrict__ cls_w,  // (10,1024)
                           const float* __restrict__ cls_b,  // (10,)
                           float* __restrict__ out)          // (256,10)
{
    const int lane = threadIdx.x & 31;
    const int wave = threadIdx.x >> 5;
    const int b    = blockIdx.x * 8 + wave;

    float part[10];
    #pragma unroll
    for (int c = 0; c < 10; ++c) part[c] = 0.f;

    const float* f = flat + (size_t)b * 1024;
    for (int k = lane; k < 1024; k += 32) {
        const float fv = f[k];
        #pragma unroll
        for (int c = 0; c < 10; ++c)
            part[c] += fv * cls_w[c * 1024 + k];
    }
    #pragma unroll
    for (int c = 0; c < 10; ++c) {
        #pragma unroll
        for (int off = 16; off > 0; off >>= 1)
            part[c] += __shfl_xor(part[c], off, 32);
        part[c] += cls_b[c];
    }

    float mx = part[0];
    #pragma unroll
    for (int c = 1; c < 10; ++c) mx = fmaxf(mx, part[c]);
    float sum = 0.f;
    #pragma unroll
    for (int c = 0; c < 10; ++c) sum += expf(part[c] - mx);
    const float lse = logf(sum) + mx;
    #pragma unroll
    for (int c = 0; c < 10; ++c)
        if (lane == c) out[b * 10 + c] = part[c] - lse;
}

// ---------------------------------------------------------------------------
extern "C" void kernel_launch(void* const* d_in, const int* in_sizes, int n_in,
                              void* d_out, int out_size, void* d_ws, size_t ws_size,
                              hipStream_t stream)
{
    const float* x     = (const float*)d_in[0];
    const float* theta = (const float*)d_in[1];
    const float* Wp    = (const float*)d_in[2];
    const float* bp    = (const float*)d_in[3];
    const float* in_w  = (const float*)d_in[4];
    const float* in_b  = (const float*)d_in[5];
    const float* out_w = (const float*)d_in[6];
    const float* out_b = (const float*)d_in[7];
    const float* cls_w = (const float*)d_in[8];
    const float* cls_b = (const float*)d_in[9];
    float* out = (float*)d_out;

    // Workspace layout (floats): ~6.1 MB total
    float* ws    = (float*)d_ws;
    float* qfeat = ws;                     // 1024*16
    float* emb   = qfeat + 1024 * 16;      // 1024*256
    float* qkv   = emb   + 1024 * 256;     // 1024*768
    float* oatt  = qkv   + 1024 * 768;     // 1024*256
    float* oproj = oatt  + 1024 * 256;     // 1024*256

    // GF(2) CNOT-circuit matrix (rows as column bitmasks), computed on host:
    // A = I; repeat 4x { A[i+1]^=A[i] fwd; A[i]^=A[i+1] bwd; }
    Masks16 masks;
    {
        unsigned int a[16];
        for (int i = 0; i < 16; ++i) a[i] = 1u << i;
        for (int t = 0; t < 4; ++t) {
            for (int i = 0; i < 15; ++i) a[i + 1] ^= a[i];
            for (int i = 0; i < 15; ++i) a[i] ^= a[i + 1];
        }
        for (int i = 0; i < 16; ++i) masks.m[i] = a[i];
    }

    // 1) quantum features (1024 x 16)
    qfeat_kernel<<<64, 256, 0, stream>>>(x, theta, qfeat, masks);
    // 2) emb = 2*(qfeat @ Wp^T + bp)   : M=1024 N=256 K=16  (alpha=2 folds emb+emb)
    gemm_wmma<<<(1024 / 16) * (256 / 16), 32, 0, stream>>>(qfeat, Wp, bp, emb, 256, 16, 2.0f);
    // 3) qkv = emb @ in_w^T + in_b     : M=1024 N=768 K=256
    gemm_wmma<<<(1024 / 16) * (768 / 16), 32, 0, stream>>>(emb, in_w, in_b, qkv, 768, 256, 1.0f);
    // 4) attention (B=256, H=8, T=4, D=32)
    attn_kernel<<<256, 256, 0, stream>>>(qkv, oatt);
    // 5) oproj = oatt @ out_w^T + out_b : M=1024 N=256 K=256
    gemm_wmma<<<(1024 / 16) * (256 / 16), 32, 0, stream>>>(oatt, out_w, out_b, oproj, 256, 256, 1.0f);
    // 6) logits + log_softmax -> d_out (256 x 10)
    cls_kernel<<<32, 256, 0, stream>>>(oproj, cls_w, cls_b, out);
}